// TransitionDown_89043261980757
// MI455X (gfx1250) — compile-verified
//
#include <hip/hip_runtime.h>
#include <hip/hip_bf16.h>

// ---------------------------------------------------------------------------
// TransitionDown (PointNet++): FPS -> kNN(16) -> gather(rel xyz + feat)
//   -> Linear(67->128, WMMA f32 16x16x4) -> BN(batch stats) -> ReLU -> maxpool
// MI455X notes: all intermediates (~63MB) are L2-resident (192MB L2), so the
// work is latency/compute bound, not HBM bound. GEMM uses V_WMMA_F32_16X16X4_F32
// to keep f32 reference precision (GEMM is only ~1.4 GFLOP, not the bottleneck).
// ---------------------------------------------------------------------------

#define NPTS   20000
#define MPTS   5000          // NPTS / STRIDE
#define KNN    16
#define CIN    64
#define COUT   128
#define KFEAT  67            // 3 + CIN
#define KPAD   68            // pad to multiple of 4 for K=4 WMMA chunks
#define BN_EPS 1e-5f

typedef float v2f __attribute__((ext_vector_type(2)));
typedef float v8f __attribute__((ext_vector_type(8)));

// ---------------------------------------------------------------------------
// 1) FPS: single workgroup, 1024 threads (32 wave32s). dist[] lives in ws
//    (80KB, L2 resident). Per iteration: strided min-update + argmax via
//    wave32 shuffle reduction then one 32-wide cross-wave LDS round.
//    Writes fidx[] to ws and n_p rows directly into d_out.
// ---------------------------------------------------------------------------
__global__ __launch_bounds__(1024) void fps_kernel(const float* __restrict__ p,
                                                   float* __restrict__ dist,
                                                   int* __restrict__ fidx,
                                                   float* __restrict__ n_p) {
    __shared__ float swval[32];
    __shared__ int   swidx[32];
    __shared__ int   s_last;
    const int tid = threadIdx.x;

    for (int j = tid; j < NPTS; j += 1024) dist[j] = 1e10f;
    if (tid == 0) {
        fidx[0] = 0;
        n_p[0] = p[0]; n_p[1] = p[1]; n_p[2] = p[2];
        s_last = 0;
    }
    __syncthreads();

    int last = 0;
    for (int i = 1; i < MPTS; ++i) {
        const float lx = p[last * 3 + 0];
        const float ly = p[last * 3 + 1];
        const float lz = p[last * 3 + 2];
        float bv = -1.0f; int bi = 0x7fffffff;
        for (int j = tid; j < NPTS; j += 1024) {
            float dx = p[j * 3 + 0] - lx;
            float dy = p[j * 3 + 1] - ly;
            float dz = p[j * 3 + 2] - lz;
            float d  = fminf(dist[j], dx * dx + dy * dy + dz * dz);
            dist[j] = d;
            if (d > bv) { bv = d; bi = j; }   // strict '>' keeps lowest index
        }
        // wave32 argmax reduction (prefer lower index on ties, like jnp.argmax)
        #pragma unroll
        for (int off = 16; off > 0; off >>= 1) {
            float ov = __shfl_down(bv, off, 32);
            int   oi = __shfl_down(bi, off, 32);
            if (ov > bv || (ov == bv && oi < bi)) { bv = ov; bi = oi; }
        }
        if ((tid & 31) == 0) { swval[tid >> 5] = bv; swidx[tid >> 5] = bi; }
        __syncthreads();
        if (tid < 32) {
            bv = swval[tid]; bi = swidx[tid];
            #pragma unroll
            for (int off = 16; off > 0; off >>= 1) {
                float ov = __shfl_down(bv, off, 32);
                int   oi = __shfl_down(bi, off, 32);
                if (ov > bv || (ov == bv && oi < bi)) { bv = ov; bi = oi; }
            }
            if (tid == 0) {
                s_last = bi;
                fidx[i] = bi;
                n_p[i * 3 + 0] = p[bi * 3 + 0];
                n_p[i * 3 + 1] = p[bi * 3 + 1];
                n_p[i * 3 + 2] = p[bi * 3 + 2];
            }
        }
        __syncthreads();
        last = s_last;
    }
}

// ---------------------------------------------------------------------------
// 2) kNN: one thread per query; all 32 lanes of a wave scan the same p[j]
//    (address-broadcast friendly). Register-resident fully-unrolled top-16
//    (single bubble pass per accepted candidate => no scratch). Prefetch hint
//    emits global_prefetch_b8.
// ---------------------------------------------------------------------------
__global__ __launch_bounds__(256) void knn_kernel(const float* __restrict__ p,
                                                  const float* __restrict__ q,
                                                  int* __restrict__ nidx) {
    const int m = blockIdx.x * blockDim.x + threadIdx.x;
    if (m >= MPTS) return;
    const float qx = q[m * 3 + 0], qy = q[m * 3 + 1], qz = q[m * 3 + 2];

    float bd[KNN];
    int   bi[KNN];
    #pragma unroll
    for (int t = 0; t < KNN; ++t) { bd[t] = 3.4e38f; bi[t] = 0; }

    for (int j = 0; j < NPTS; ++j) {
        if (j + 128 < NPTS) __builtin_prefetch(&p[(j + 128) * 3], 0, 0);
        float dx = p[j * 3 + 0] - qx;
        float dy = p[j * 3 + 1] - qy;
        float dz = p[j * 3 + 2] - qz;
        float d  = dx * dx + dy * dy + dz * dz;
        if (d < bd[KNN - 1]) {
            bd[KNN - 1] = d; bi[KNN - 1] = j;
            #pragma unroll
            for (int t = KNN - 1; t > 0; --t) {   // constant indices: registers
                if (bd[t] < bd[t - 1]) {
                    float tf = bd[t]; bd[t] = bd[t - 1]; bd[t - 1] = tf;
                    int   ti = bi[t]; bi[t] = bi[t - 1]; bi[t - 1] = ti;
                }
            }
        }
    }
    #pragma unroll
    for (int t = 0; t < KNN; ++t) nidx[m * KNN + t] = bi[t];
}

// ---------------------------------------------------------------------------
// 3) Gather: build feat [M*KNN, KPAD] row-major; rel xyz | x[nidx] | zero pad.
// ---------------------------------------------------------------------------
__global__ __launch_bounds__(256) void gather_kernel(const float* __restrict__ p,
                                                     const float* __restrict__ x,
                                                     const float* __restrict__ n_p,
                                                     const int* __restrict__ nidx,
                                                     float* __restrict__ feat) {
    const int r = blockIdx.x * blockDim.x + threadIdx.x;
    if (r >= MPTS * KNN) return;
    const int m = r / KNN;
    const int j = nidx[r];
    float* f = feat + (size_t)r * KPAD;
    f[0] = p[j * 3 + 0] - n_p[m * 3 + 0];
    f[1] = p[j * 3 + 1] - n_p[m * 3 + 1];
    f[2] = p[j * 3 + 2] - n_p[m * 3 + 2];
    #pragma unroll
    for (int c = 0; c < CIN; ++c) f[3 + c] = x[(size_t)j * CIN + c];
    f[3 + CIN] = 0.0f;   // K pad (67 -> 68)
}

// ---------------------------------------------------------------------------
// 4) GEMM h = feat @ W with V_WMMA_F32_16X16X4_F32.
//    Block = 256 threads = 8 wave32s. One block per 16-row tile (5000 blocks);
//    wave w owns the 16x16 tile at cols [16w, 16w+16). A tile + zero-padded W
//    staged in LDS (~40KB of 320KB WGP LDS). Per-block BN partial sums via
//    LDS ds_add_f32 atomics, flushed with one global_atomic_add_f32 per chan.
//    A frag layout (16x4 f32): lane l<16 -> M=l, {K0,K1}; lane>=16 -> {K2,K3}.
//    B frag layout (4x16 f32): VGPR0 rows {K0,K2}, VGPR1 rows {K1,K3} by half.
// ---------------------------------------------------------------------------
__global__ __launch_bounds__(256) void gemm_kernel(const float* __restrict__ feat,
                                                   const float* __restrict__ Wg,
                                                   float* __restrict__ h,
                                                   float* __restrict__ stats) {
    __shared__ float As[16][KPAD];       // 4352 B
    __shared__ float Ws[KPAD][COUT];     // 34816 B
    __shared__ float bsum[COUT];
    __shared__ float bsq[COUT];

    const int tid  = threadIdx.x;
    const int tile = blockIdx.x;         // 0..4999

    for (int i = tid; i < KPAD * COUT; i += 256) {
        int k = i / COUT, c = i % COUT;
        Ws[k][c] = (k < KFEAT) ? Wg[k * COUT + c] : 0.0f;
    }
    for (int i = tid; i < 16 * KPAD; i += 256) {
        int rr = i / KPAD, k = i % KPAD;
        As[rr][k] = feat[(size_t)(tile * 16 + rr) * KPAD + k];
    }
    for (int i = tid; i < COUT; i += 256) { bsum[i] = 0.0f; bsq[i] = 0.0f; }
    __syncthreads();

    const int wave = tid >> 5;           // column tile 0..7
    const int lane = tid & 31;
    const int half = lane >> 4;          // 0 or 1
    const int l16  = lane & 15;
    const int col  = wave * 16 + l16;

    v8f c = {};
    #pragma unroll
    for (int kc = 0; kc < KPAD / 4; ++kc) {     // 17 chunks of K=4
        const int kb = kc * 4 + half * 2;
        v2f a, b;
        a.x = As[l16][kb];
        a.y = As[l16][kb + 1];
        b.x = Ws[kb][col];
        b.y = Ws[kb + 1][col];
        c = __builtin_amdgcn_wmma_f32_16x16x4_f32(
                /*neg_a=*/false, a, /*neg_b=*/false, b,
                /*c_mod=*/(short)0, c, /*reuse_a=*/false, /*reuse_b=*/false);
    }

    // D layout: VGPR v -> M = v + 8*half, N = l16. Store h + local BN sums.
    float s = 0.0f, sq = 0.0f;
    #pragma unroll
    for (int v = 0; v < 8; ++v) {
        const int row = tile * 16 + half * 8 + v;
        const float val = c[v];
        h[(size_t)row * COUT + col] = val;
        s  += val;
        sq += val * val;
    }
    atomicAdd(&bsum[col], s);            // ds_add_f32
    atomicAdd(&bsq[col], sq);
    __syncthreads();
    if (tid < COUT) {
        atomicAdd(&stats[tid], bsum[tid]);          // global_atomic_add_f32
        atomicAdd(&stats[COUT + tid], bsq[tid]);
    }
}

// ---------------------------------------------------------------------------
// 5) Finalize: BN (batch stats over M*KNN), ReLU, maxpool over KNN -> x_out.
//    One thread per (m, channel). Also writes n_o as int32.
// ---------------------------------------------------------------------------
__global__ __launch_bounds__(256) void finalize_kernel(const float* __restrict__ h,
                                                       const float* __restrict__ stats,
                                                       const float* __restrict__ gamma,
                                                       const float* __restrict__ beta,
                                                       float* __restrict__ x_out,
                                                       int* __restrict__ n_o) {
    const int i = blockIdx.x * blockDim.x + threadIdx.x;
    if (i == 0) n_o[0] = MPTS;
    if (i >= MPTS * COUT) return;
    const int m   = i / COUT;
    const int col = i % COUT;

    const float cnt   = (float)(MPTS * KNN);
    const float mean  = stats[col] / cnt;
    const float var   = stats[COUT + col] / cnt - mean * mean;
    const float scale = gamma[col] * rsqrtf(var + BN_EPS);
    const float shift = beta[col] - mean * scale;

    float best = 0.0f;                   // relu(x) >= 0, so 0 is the identity
    #pragma unroll
    for (int s = 0; s < KNN; ++s) {
        float v = h[(size_t)(m * KNN + s) * COUT + col];
        v = fmaxf(v * scale + shift, 0.0f);
        best = fmaxf(best, v);
    }
    x_out[i] = best;
}

// ---------------------------------------------------------------------------
// Launch. Inputs: p[60000] x[1280000] o[1] W[8576] gamma[128] beta[128].
// d_out: n_p (15000 f32) | x_out (640000 f32) | n_o (1 i32).
// Workspace layout (bytes):
//   dist   @ 0          : 20000 * 4      =    80,000
//   nidx   @ 80000      : 5000*16*4      =   320,000
//   stats  @ 400000     : 256 * 4        =     1,024
//   fidx   @ 401024     : 5000 * 4       =    20,000  (pad to 421,376)
//   feat   @ 421376     : 80000*68*4     = 21,760,000
//   h      @ 22,181,376 : 80000*128*4    = 40,960,000   (total ~63.2 MB)
// ---------------------------------------------------------------------------
extern "C" void kernel_launch(void* const* d_in, const int* in_sizes, int n_in,
                              void* d_out, int out_size, void* d_ws, size_t ws_size,
                              hipStream_t stream) {
    const float* p     = (const float*)d_in[0];
    const float* x     = (const float*)d_in[1];
    const float* Wg    = (const float*)d_in[3];
    const float* gamma = (const float*)d_in[4];
    const float* beta  = (const float*)d_in[5];
    (void)in_sizes; (void)n_in; (void)out_size;

    char* ws = (char*)d_ws;
    float* dist  = (float*)(ws + 0);
    int*   nidx  = (int*)  (ws + 80000);
    float* stats = (float*)(ws + 400000);
    int*   fidx  = (int*)  (ws + 401024);
    float* feat  = (float*)(ws + 421376);
    float* h     = (float*)(ws + 22181376);
    (void)ws_size;

    float* n_p   = (float*)d_out;                       // [M,3]
    float* x_out = (float*)d_out + (size_t)MPTS * 3;    // [M,COUT]
    int*   n_o   = (int*)((float*)d_out + (size_t)MPTS * 3 + (size_t)MPTS * COUT);

    hipMemsetAsync(stats, 0, 2 * COUT * sizeof(float), stream);

    fps_kernel<<<1, 1024, 0, stream>>>(p, dist, fidx, n_p);
    knn_kernel<<<(MPTS + 255) / 256, 256, 0, stream>>>(p, n_p, nidx);
    gather_kernel<<<(MPTS * KNN + 255) / 256, 256, 0, stream>>>(p, x, n_p, nidx, feat);
    gemm_kernel<<<MPTS, 256, 0, stream>>>(feat, Wg, h, stats);
    finalize_kernel<<<(MPTS * COUT + 255) / 256, 256, 0, stream>>>(h, stats, gamma, beta, x_out, n_o);
}